// AttentionPool_v1_25168508355180
// MI455X (gfx1250) — compile-verified
//
#include <hip/hip_runtime.h>
#include <math.h>

// Problem constants (from reference)
#define NB    16
#define CC    64
#define WW    512
#define HH    512
#define REDD  64
#define OUTW  128
#define BN_EPS 1e-5f

typedef __attribute__((ext_vector_type(2))) float v2f;
typedef __attribute__((ext_vector_type(8))) float v8f;

// ---------------------------------------------------------------------------
// Kernel 1: pooled[n][w] = max over (c,h) of x[n,c,w,h].  Reads 1 GiB.
// One block per (n,w): 64 rows of 512 contiguous floats = 8192 float4.
// ---------------------------------------------------------------------------
__global__ __launch_bounds__(256) void pool_max_kernel(
    const float* __restrict__ x, float* __restrict__ pooled) {
  int b = blockIdx.x;            // n*512 + w
  int n = b >> 9;
  int w = b & 511;
  const float* base = x + (size_t)n * CC * WW * HH + (size_t)w * HH;
  int t = threadIdx.x;
  float m = -INFINITY;
  for (int f = t; f < CC * (HH / 4); f += 256) {   // 32 float4 per thread
    int c  = f >> 7;                               // f / 128
    int h4 = f & 127;
    float4 v = *((const float4*)(base + (size_t)c * WW * HH) + h4);
    m = fmaxf(m, fmaxf(fmaxf(v.x, v.y), fmaxf(v.z, v.w)));
  }
  __shared__ float red[256];
  red[t] = m;
  __syncthreads();
  for (int s = 128; s > 0; s >>= 1) {
    if (t < s) red[t] = fmaxf(red[t], red[t + s]);
    __syncthreads();
  }
  if (t == 0) pooled[b] = red[0];
}

// ---------------------------------------------------------------------------
// Kernel 2: h = relu(BN(pooled @ w1.T + b1)); logits = h @ w2.T + b2
// 1 block, 128 threads = 4 wave32; fp32 WMMA 16x16x4 for both GEMMs.
// softmax skipped: monotonic, does not change top-k indices.
// ---------------------------------------------------------------------------
__global__ __launch_bounds__(128) void mlp_kernel(
    const float* __restrict__ pooled, const float* __restrict__ w1,
    const float* __restrict__ b1,     const float* __restrict__ gamma,
    const float* __restrict__ beta,   const float* __restrict__ w2,
    const float* __restrict__ b2,     float* __restrict__ logits) {
  __shared__ float sH[16][REDD + 4];          // +4 pad: conflict-free column reads

  const int tid  = threadIdx.x;
  const int lane = tid & 31;
  const int wv   = tid >> 5;                  // wave id 0..3
  const int m    = lane & 15;                 // matrix row this lane holds
  const int kh   = (lane >> 4) << 1;          // K sub-offset: 0 or 2
  const int rbase = (lane >> 4) << 3;         // C/D: rows r (+0) or r+8

  // ---- GEMM1: h[16][64] = pooled[16][512] @ w1.T ; wave wv owns cols 16wv..+15
  {
    const int col = (wv << 4) + m;
    const float* arow = pooled + m * WW;      // A[m][k]
    const float* brow = w1 + col * WW;        // B[k][col] = w1[col][k]
    v8f acc = {};
#pragma unroll 8
    for (int k0 = 0; k0 < WW; k0 += 4) {
      v2f a = *(const v2f*)(arow + k0 + kh);  // A[m][k0+kh], A[m][k0+kh+1]
      v2f bb = *(const v2f*)(brow + k0 + kh); // B[k0+kh][col], B[k0+kh+1][col]
      acc = __builtin_amdgcn_wmma_f32_16x16x4_f32(
          false, a, false, bb, (short)0, acc, false, false);
    }
    float bias = b1[col];
#pragma unroll
    for (int r = 0; r < 8; ++r) sH[r + rbase][col] = acc[r] + bias;
  }
  __syncthreads();

  // ---- BatchNorm1d (training stats over batch of 16) + ReLU, per feature
  if (tid < REDD) {
    float mu = 0.f;
#pragma unroll
    for (int r = 0; r < 16; ++r) mu += sH[r][tid];
    mu *= (1.0f / 16.0f);
    float var = 0.f;
#pragma unroll
    for (int r = 0; r < 16; ++r) { float d = sH[r][tid] - mu; var += d * d; }
    var *= (1.0f / 16.0f);
    float inv = rsqrtf(var + BN_EPS);
    float g = gamma[tid], be = beta[tid];
#pragma unroll
    for (int r = 0; r < 16; ++r) {
      float v = g * (sH[r][tid] - mu) * inv + be;
      sH[r][tid] = v > 0.f ? v : 0.f;
    }
  }
  __syncthreads();

  // ---- GEMM2: logits[16][512] = h[16][64] @ w2.T ; 8 16-col tiles per wave
  for (int tile = wv; tile < WW / 16; tile += 4) {
    const int col = (tile << 4) + m;
    const float* brow = w2 + col * REDD;      // B[k][col] = w2[col][k]
    v8f acc = {};
#pragma unroll
    for (int k0 = 0; k0 < REDD; k0 += 4) {
      v2f a;
      a.x = sH[m][k0 + kh];
      a.y = sH[m][k0 + kh + 1];
      v2f bb = *(const v2f*)(brow + k0 + kh);
      acc = __builtin_amdgcn_wmma_f32_16x16x4_f32(
          false, a, false, bb, (short)0, acc, false, false);
    }
    float bias = b2[col];
#pragma unroll
    for (int r = 0; r < 8; ++r) logits[(r + rbase) * WW + col] = acc[r] + bias;
  }
}

// ---------------------------------------------------------------------------
// Kernel 3: per row, select indices with rank < 128 (value desc, tie -> lower
// index, matching jax.lax.top_k), emit in ascending index order.
// 16 blocks x 512 threads; ranks form a permutation of 0..511.
// ---------------------------------------------------------------------------
__global__ __launch_bounds__(512) void topk_kernel(
    const float* __restrict__ logits, int* __restrict__ idx) {
  __shared__ float row[WW];
  __shared__ unsigned char sel[WW];
  int n = blockIdx.x;
  int j = threadIdx.x;
  row[j] = logits[n * WW + j];
  __syncthreads();
  float v = row[j];
  int rank = 0;
  for (int k = 0; k < WW; ++k) {               // LDS broadcast reads
    float u = row[k];
    rank += (u > v) || (u == v && k < j);
  }
  sel[j] = (rank < OUTW) ? 1 : 0;
  __syncthreads();
  if (rank < OUTW) {
    int pos = 0;
    for (int k = 0; k < j; ++k) pos += sel[k]; // position among ascending picks
    idx[n * OUTW + pos] = j;
  }
}

// ---------------------------------------------------------------------------
// Kernel 4: out[n,c,o,:] = x[n,c,idx[n][o],:].  256 MiB read + 256 MiB write.
// One block per (n,c); idx row cached in LDS; float4 coalesced copies.
// ---------------------------------------------------------------------------
__global__ __launch_bounds__(256) void gather_kernel(
    const float* __restrict__ x, const int* __restrict__ idx,
    float* __restrict__ out) {
  __shared__ int sidx[OUTW];
  int b = blockIdx.x;                          // n*64 + c
  int n = b >> 6;
  int c = b & 63;
  int t = threadIdx.x;
  if (t < OUTW) sidx[t] = idx[n * OUTW + t];
  __syncthreads();
  const float* src_base = x   + (size_t)n * CC * WW   * HH + (size_t)c * WW   * HH;
  float*       dst_base = out + (size_t)n * CC * OUTW * HH + (size_t)c * OUTW * HH;
  for (int f = t; f < OUTW * (HH / 4); f += 256) {   // 64 float4 per thread
    int o  = f >> 7;
    int h4 = f & 127;
    float4 v = *((const float4*)(src_base + (size_t)sidx[o] * HH) + h4);
    *((float4*)(dst_base + (size_t)o * HH) + h4) = v;
  }
}

// ---------------------------------------------------------------------------
extern "C" void kernel_launch(void* const* d_in, const int* in_sizes, int n_in,
                              void* d_out, int out_size, void* d_ws, size_t ws_size,
                              hipStream_t stream) {
  const float* x     = (const float*)d_in[0];
  const float* w1    = (const float*)d_in[1];
  const float* b1    = (const float*)d_in[2];
  const float* gamma = (const float*)d_in[3];
  const float* beta  = (const float*)d_in[4];
  const float* w2    = (const float*)d_in[5];
  const float* b2    = (const float*)d_in[6];
  float* out = (float*)d_out;

  char* ws = (char*)d_ws;
  float* pooled = (float*)ws;                       // 16*512 f32 = 32 KiB
  float* logits = (float*)(ws + 8192 * 4);          // 16*512 f32 = 32 KiB
  int*   idx    = (int*)  (ws + 2 * 8192 * 4);      // 16*128 i32 =  8 KiB

  pool_max_kernel<<<NB * WW, 256, 0, stream>>>(x, pooled);
  mlp_kernel<<<1, 128, 0, stream>>>(pooled, w1, b1, gamma, beta, w2, b2, logits);
  topk_kernel<<<NB, WW, 0, stream>>>(logits, idx);
  gather_kernel<<<NB * CC, 256, 0, stream>>>(x, idx, out);
}